// EnsembleModel_41970420417941
// MI455X (gfx1250) — compile-verified
//
#include <hip/hip_runtime.h>

// ---------------------------------------------------------------------------
// EnsembleModel for MI455X (gfx1250, wave32).
//
// Cost model: GEMMs are ~0.8 GFLOP (trivial). Dominant traffic is the
// 2 x 512 x 100000 f32 ratings write (~410 MB) -> HBM-store bound (~20us
// at 23.3 TB/s). So: nontemporal vector zero-fill, WMMA f32 16x16x4 for the
// two GEMM stages (one wave per 16x16 tile), scatter+mask fused into the
// GEMM epilogue, LDS-resident iterative top-20 per row.
// ---------------------------------------------------------------------------

#define N_ITEMS   100000
#define B_ROWS    512
#define D_LAT     64
#define K_TOP     20
#define N_SMALL   2000
#define N_MID     10000

typedef __attribute__((ext_vector_type(2))) float v2f;
typedef __attribute__((ext_vector_type(4))) float v4f;
typedef __attribute__((ext_vector_type(8))) float v8f;

// ---------------- zero fill (nontemporal, 16B per lane) --------------------
__global__ void fill_zero_kernel(float* __restrict__ out, size_t n_floats) {
    size_t i      = (size_t)blockIdx.x * blockDim.x + threadIdx.x;
    size_t stride = (size_t)gridDim.x * blockDim.x;
    size_t n4     = n_floats >> 2;
    v4f z = {0.f, 0.f, 0.f, 0.f};
    v4f* p = (v4f*)out;
    for (size_t j = i; j < n4; j += stride)
        __builtin_nontemporal_store(z, p + j);
}

// ---------------- stage 1: H = X @ prior_W  (both branches) ----------------
// One wave per 16x16 tile of H. K=64 -> 16 x V_WMMA_F32_16X16X4_F32.
// A 16x4 f32 layout: lanes 0-15 -> K={k0,k0+1}, lanes 16-31 -> K={k0+2,k0+3},
// M = lane&15. B 4x16 mirrored with N = lane&15. D: M = v + 8*(lane>=16).
__global__ __launch_bounds__(32)
void gemm_h_kernel(const float* __restrict__ X,   // [512][64]
                   const float* __restrict__ Ws,  // [64][64]
                   const float* __restrict__ Wm,  // [64][64]
                   float* __restrict__ H) {       // [2][512][64]
    const int lane   = threadIdx.x;      // 0..31
    const int tile_n = blockIdx.x;       // 0..3
    const int tile_m = blockIdx.y;       // 0..31
    const int br     = blockIdx.z;       // 0..1
    const float* W   = br ? Wm : Ws;
    float* Hout      = H + (size_t)br * B_ROWS * D_LAT;

    const int r  = lane & 15;
    const int kh = (lane >> 4) << 1;     // 0 or 2

    v8f acc = {};
#pragma unroll
    for (int k0 = 0; k0 < D_LAT; k0 += 4) {
        const int ka = k0 + kh;
        v2f a = *(const v2f*)(X + (size_t)(tile_m * 16 + r) * D_LAT + ka);
        v2f b;
        b.x = W[(size_t)(ka    ) * D_LAT + tile_n * 16 + r];
        b.y = W[(size_t)(ka + 1) * D_LAT + tile_n * 16 + r];
        acc = __builtin_amdgcn_wmma_f32_16x16x4_f32(
            /*neg_a=*/false, a, /*neg_b=*/false, b,
            /*c_mod=*/(short)0, acc, /*reuse_a=*/false, /*reuse_b=*/false);
    }

    const int mbase = (lane >> 4) << 3;  // 0 or 8
#pragma unroll
    for (int v = 0; v < 8; ++v) {
        const int gr = tile_m * 16 + v + mbase;
        Hout[(size_t)gr * D_LAT + tile_n * 16 + r] = acc[v];
    }
}

// ------- stage 2: preds = H @ dec_W, scatter*mask into ratings -------------
__global__ __launch_bounds__(32)
void gemm_scatter_kernel(const float* __restrict__ H,    // [512][64]
                         const float* __restrict__ Wd,   // [64][n]
                         const int*   __restrict__ map,  // [n]
                         const float* __restrict__ mask, // [512][N_ITEMS]
                         float* __restrict__ out,        // [512][N_ITEMS]
                         int n) {
    const int lane   = threadIdx.x;
    const int tile_n = blockIdx.x;
    const int tile_m = blockIdx.y;
    const int r  = lane & 15;
    const int kh = (lane >> 4) << 1;

    v8f acc = {};
#pragma unroll
    for (int k0 = 0; k0 < D_LAT; k0 += 4) {
        const int ka = k0 + kh;
        v2f a = *(const v2f*)(H + (size_t)(tile_m * 16 + r) * D_LAT + ka);
        v2f b;
        b.x = Wd[(size_t)(ka    ) * n + tile_n * 16 + r];
        b.y = Wd[(size_t)(ka + 1) * n + tile_n * 16 + r];
        acc = __builtin_amdgcn_wmma_f32_16x16x4_f32(
            false, a, false, b, (short)0, acc, false, false);
    }

    const int j    = tile_n * 16 + r;    // subset column this lane owns
    const int item = map[j];             // injective -> no write conflicts
    const int mbase = (lane >> 4) << 3;
#pragma unroll
    for (int v = 0; v < 8; ++v) {
        const int gr = tile_m * 16 + v + mbase;
        const float mv = mask[(size_t)gr * N_ITEMS + item];
        out[(size_t)gr * N_ITEMS + item] = acc[v] * mv;
    }
}

// ---------------- stage 3: per-row iterative top-20 ------------------------
// One 256-thread workgroup per row. Candidate set = the n scattered values
// (everything else is exactly 0). Values staged in LDS (max 10000 f32 =
// 40 KB, well under the 320 KB WGP LDS); 20 rounds of argmax with the winner
// knocked out to -inf. Ties break toward the smaller subset index.
__global__ __launch_bounds__(256)
void topk_kernel(const float* __restrict__ ratings, // [512][N_ITEMS]
                 const int*   __restrict__ map,     // [n]
                 int n,
                 float* __restrict__ topk_out) {    // [512][K_TOP] (as f32)
    __shared__ float vals[N_MID];
    __shared__ float sval[256];
    __shared__ int   sidx[256];

    const int row = blockIdx.x;
    const int tid = threadIdx.x;
    const float* rrow = ratings + (size_t)row * N_ITEMS;

    for (int j = tid; j < n; j += 256)
        vals[j] = rrow[map[j]];
    __syncthreads();

    for (int i = 0; i < K_TOP; ++i) {
        float bv = -__builtin_inff();
        int   bj = n;
        for (int j = tid; j < n; j += 256) {
            const float v = vals[j];
            if (v > bv || (v == bv && j < bj)) { bv = v; bj = j; }
        }
        sval[tid] = bv;
        sidx[tid] = bj;
        __syncthreads();
#pragma unroll
        for (int s = 128; s > 0; s >>= 1) {
            if (tid < s) {
                const float ov = sval[tid + s];
                const int   oj = sidx[tid + s];
                if (ov > sval[tid] || (ov == sval[tid] && oj < sidx[tid])) {
                    sval[tid] = ov;
                    sidx[tid] = oj;
                }
            }
            __syncthreads();
        }
        if (tid == 0) {
            const int jm = sidx[0];
            topk_out[(size_t)row * K_TOP + i] = (float)map[jm];
            vals[jm] = -__builtin_inff();
        }
        __syncthreads();
    }
}

// ---------------------------------------------------------------------------
extern "C" void kernel_launch(void* const* d_in, const int* in_sizes, int n_in,
                              void* d_out, int out_size, void* d_ws, size_t ws_size,
                              hipStream_t stream) {
    const float* X    = (const float*)d_in[0];
    const float* sW   = (const float*)d_in[1];   // small prior [64][64]
    const float* sD   = (const float*)d_in[2];   // small dec   [64][2000]
    const float* mW   = (const float*)d_in[3];   // mid prior   [64][64]
    const float* mD   = (const float*)d_in[4];   // mid dec     [64][10000]
    const int*   tmap = (const int*)d_in[5];     // [2000]
    const int*   mmap = (const int*)d_in[6];     // [10000]
    const float* mask = (const float*)d_in[7];   // [512][100000]
    // d_in[8] = k (==20), hardcoded.

    float* out   = (float*)d_out;
    float* rat_s = out;
    float* rat_m = out + (size_t)B_ROWS * N_ITEMS;
    float* top_s = out + 2 * (size_t)B_ROWS * N_ITEMS;
    float* top_m = top_s + (size_t)B_ROWS * K_TOP;

    float* H = (float*)d_ws;                     // [2][512][64] = 256 KB

    // 1) zero both ratings planes (the bandwidth-dominant step).
    const size_t nfill = 2 * (size_t)B_ROWS * N_ITEMS;
    fill_zero_kernel<<<4096, 256, 0, stream>>>(out, nfill);

    // 2) H = X @ prior_W for both branches (WMMA f32).
    gemm_h_kernel<<<dim3(D_LAT / 16, B_ROWS / 16, 2), 32, 0, stream>>>(X, sW, mW, H);

    // 3) preds = H @ dec_W, scatter * mask into the ratings planes.
    gemm_scatter_kernel<<<dim3(N_SMALL / 16, B_ROWS / 16), 32, 0, stream>>>(
        H, sD, tmap, mask, rat_s, N_SMALL);
    gemm_scatter_kernel<<<dim3(N_MID / 16, B_ROWS / 16), 32, 0, stream>>>(
        H + (size_t)B_ROWS * D_LAT, mD, mmap, mask, rat_m, N_MID);

    // 4) top-20 per row per branch (indices emitted as f32 values).
    topk_kernel<<<B_ROWS, 256, 0, stream>>>(rat_s, tmap, N_SMALL, top_s);
    topk_kernel<<<B_ROWS, 256, 0, stream>>>(rat_m, mmap, N_MID, top_m);
}